// SimpleMoE_47949014892589
// MI455X (gfx1250) — compile-verified
//
#include <hip/hip_runtime.h>

typedef __attribute__((ext_vector_type(2))) float v2f;
typedef __attribute__((ext_vector_type(8))) float v8f;

constexpr int E_ = 8;     // experts
constexpr int D_ = 1024;  // model dim
constexpr int H_ = 4096;  // hidden dim
constexpr int MT = 64;    // token tile
constexpr int HB = 128;   // hidden block per fused step
constexpr int XS = D_ + 4;  // padded LDS row stride for X (bank-conflict-free)
constexpr int HS = HB + 4;  // padded LDS row stride for H
constexpr int NTHREADS = 512;  // 16 waves (wave32)

// ---------------------------------------------------------------------------
// Router: logits = x @ rw + rb ; softmax ; top-2 ; renormalized gates ;
// scatter token indices + gates into per-expert compacted lists.
// One wave32 per token.
// ---------------------------------------------------------------------------
__global__ void moe_router_kernel(const float* __restrict__ x,
                                  const float* __restrict__ rw,
                                  const float* __restrict__ rb,
                                  int T,
                                  int* __restrict__ counts,
                                  int* __restrict__ lists,
                                  float* __restrict__ glist) {
  int tok = (blockIdx.x * (blockDim.x >> 5)) + (threadIdx.x >> 5);
  int lane = threadIdx.x & 31;
  if (tok >= T) return;

  const float* xr = x + (size_t)tok * D_;
  float acc[E_];
#pragma unroll
  for (int e = 0; e < E_; ++e) acc[e] = 0.f;

  for (int d = lane; d < D_; d += 32) {
    float xv = xr[d];
    const float* wr = rw + (size_t)d * E_;
#pragma unroll
    for (int e = 0; e < E_; ++e) acc[e] += xv * wr[e];
  }
#pragma unroll
  for (int e = 0; e < E_; ++e) {
    for (int off = 16; off > 0; off >>= 1)
      acc[e] += __shfl_xor(acc[e], off, 32);
  }

  if (lane == 0) {
    float lg[E_];
    float m = -3.4e38f;
#pragma unroll
    for (int e = 0; e < E_; ++e) {
      lg[e] = acc[e] + rb[e];
      m = fmaxf(m, lg[e]);
    }
    float p[E_];
#pragma unroll
    for (int e = 0; e < E_; ++e) p[e] = __expf(lg[e] - m);
    // top-2 (ties -> lowest index, matching lax.top_k)
    int i1 = 0; float p1 = p[0];
#pragma unroll
    for (int e = 1; e < E_; ++e) { if (p[e] > p1) { p1 = p[e]; i1 = e; } }
    int i2 = -1; float p2 = -1.f;
#pragma unroll
    for (int e = 0; e < E_; ++e) {
      if (e != i1 && p[e] > p2) { p2 = p[e]; i2 = e; }
    }
    float inv = 1.f / (p1 + p2);   // softmax denominator cancels in the ratio
    int s1 = atomicAdd(&counts[i1], 1);
    lists[(size_t)i1 * T + s1] = tok;
    glist[(size_t)i1 * T + s1] = p1 * inv;
    int s2 = atomicAdd(&counts[i2], 1);
    lists[(size_t)i2 * T + s2] = tok;
    glist[(size_t)i2 * T + s2] = p2 * inv;
  }
}

// ---------------------------------------------------------------------------
// Fused expert MLP: out[tok] += gate * (relu(x@w1+b1) @ w2 + b2)
// One block = (expert e, tile of up to 64 assigned tokens).
// 16 waves; GEMM1 -> LDS h-block -> GEMM2 with persistent f32 WMMA accums.
// X tile gathered via async global->LDS (ASYNCcnt); next-block weights
// prefetched into L2 while current block computes.
// ---------------------------------------------------------------------------
__global__ __launch_bounds__(NTHREADS)
void moe_expert_kernel(const float* __restrict__ x,
                       const float* __restrict__ w1,
                       const float* __restrict__ b1,
                       const float* __restrict__ w2,
                       const float* __restrict__ b2,
                       const int* __restrict__ counts,
                       const int* __restrict__ lists,
                       const float* __restrict__ glist,
                       float* __restrict__ out, int T) {
  extern __shared__ float smem[];
  float* sX = smem;            // [MT][XS]
  float* sH = smem + MT * XS;  // [MT][HS]
  __shared__ int sTok[MT];
  __shared__ float sGate[MT];

  const int e = blockIdx.y;
  const int cnt = counts[e];
  const int base = blockIdx.x * MT;
  if (base >= cnt) return;
  const int rows = min(MT, cnt - base);

  const int tid = threadIdx.x;
  const int wave = tid >> 5;
  const int lane = tid & 31;
  const int lm = lane & 15;          // N / M index within a 16-wide tile
  const int ko = (lane >> 4) << 1;   // K offset: lanes 16-31 hold K=2,3
  const int moff = (lane >> 4) << 3; // C/D: lanes 16-31 hold M=r+8

  if (tid < MT) {
    if (tid < rows) {
      sTok[tid]  = lists[(size_t)e * T + base + tid];
      sGate[tid] = glist[(size_t)e * T + base + tid];
    } else {
      sTok[tid] = 0;
      sGate[tid] = 0.f;
    }
  }
  __syncthreads();

  // Gather X tile via async global->LDS copies (16B per lane per issue).
  // Ragged tail rows are zero-filled with plain LDS stores.
  for (int idx = tid; idx < MT * (D_ / 4); idx += NTHREADS) {
    int r = idx >> 8;            // / (D_/4)
    int c = (idx & 255) << 2;    // float column
    float* ldsp = sX + (size_t)r * XS + c;
    if (r < rows) {
      unsigned lds32 = (unsigned)(unsigned long long)ldsp;
      unsigned long long gaddr =
          (unsigned long long)(x + (size_t)sTok[r] * D_ + c);
      asm volatile("global_load_async_to_lds_b128 %0, %1, off"
                   :: "v"(lds32), "v"(gaddr)
                   : "memory");
    } else {
      float4 z = {0.f, 0.f, 0.f, 0.f};
      *(float4*)ldsp = z;
    }
  }
  asm volatile("s_wait_asynccnt 0" ::: "memory");
  __syncthreads();

  const v8f zero8 = {0.f, 0.f, 0.f, 0.f, 0.f, 0.f, 0.f, 0.f};
  v8f acc2[4][4];
#pragma unroll
  for (int mt = 0; mt < 4; ++mt)
#pragma unroll
    for (int nt = 0; nt < 4; ++nt) acc2[mt][nt] = zero8;

  const int colg = wave & 7;   // GEMM1: which 16-col slice of the HB block
  const int mtg  = wave >> 3;  // GEMM1: which 32-row half of the tile
  const int n0   = wave * 64;  // GEMM2: this wave's 64 output columns
  const size_t w1e = (size_t)e * D_ * H_;
  const size_t w2e = (size_t)e * H_ * D_;

  for (int hb = 0; hb < H_ / HB; ++hb) {
    // ---- GEMM1: h[0:64, hb*HB : +HB] = relu(X @ W1 + b1) -------------------
    const int hcol = hb * HB + colg * 16 + lm;
    const float* aR0 = sX + (size_t)(mtg * 32 + lm) * XS;
    const float* aR1 = sX + (size_t)(mtg * 32 + 16 + lm) * XS;
    const float* bc  = w1 + w1e + hcol;  // column hcol, stride H_ per D-row
    v8f h0 = zero8, h1 = zero8;
    for (int k = 0; k < D_; k += 4) {
      v2f a0 = {aR0[k + ko], aR0[k + ko + 1]};
      v2f a1 = {aR1[k + ko], aR1[k + ko + 1]};
      v2f bb = {bc[(size_t)(k + ko) * H_], bc[(size_t)(k + ko + 1) * H_]};
      h0 = __builtin_amdgcn_wmma_f32_16x16x4_f32(false, a0, false, bb,
                                                 (short)0, h0, false, false);
      h1 = __builtin_amdgcn_wmma_f32_16x16x4_f32(false, a1, false, bb,
                                                 (short)0, h1, false, false);
    }

    // Prefetch next hb block of w1/w2 into L2 (overlaps barrier + GEMM2).
    if (hb + 1 < H_ / HB) {
      const float* p1 = w1 + w1e + (size_t)(hb + 1) * HB;
      for (int i = tid; i < 4096; i += NTHREADS)  // 1024 rows x 4 lines
        __builtin_prefetch(p1 + (size_t)(i >> 2) * H_ + (i & 3) * 32, 0, 1);
      const float* p2 = w2 + w2e + (size_t)((hb + 1) * HB) * D_;
      for (int i = tid; i < 4096; i += NTHREADS)  // 128 rows x 32 lines
        __builtin_prefetch(p2 + (size_t)(i >> 5) * D_ + (i & 31) * 32, 0, 1);
    }

    __syncthreads();  // previous iteration's GEMM2 done reading sH
    {
      float bias = b1[(size_t)e * H_ + hcol];
      const int cloc = colg * 16 + lm;
#pragma unroll
      for (int r = 0; r < 8; ++r) {
        float v0 = h0[r] + bias;
        sH[(size_t)(mtg * 32 + moff + r) * HS + cloc] = v0 > 0.f ? v0 : 0.f;
        float v1 = h1[r] + bias;
        sH[(size_t)(mtg * 32 + 16 + moff + r) * HS + cloc] = v1 > 0.f ? v1 : 0.f;
      }
    }
    __syncthreads();  // sH block complete

    // ---- GEMM2: acc2 += H_block @ W2[hb*HB : +HB, n0 : n0+64] --------------
    const float* w2blk = w2 + w2e + (size_t)(hb * HB) * D_;
    for (int kk = 0; kk < HB; kk += 4) {
      const float* h0p = sH + (size_t)(0 * 16 + lm) * HS + kk + ko;
      const float* h1p = sH + (size_t)(1 * 16 + lm) * HS + kk + ko;
      const float* h2p = sH + (size_t)(2 * 16 + lm) * HS + kk + ko;
      const float* h3p = sH + (size_t)(3 * 16 + lm) * HS + kk + ko;
      v2f a0 = {h0p[0], h0p[1]};
      v2f a1 = {h1p[0], h1p[1]};
      v2f a2 = {h2p[0], h2p[1]};
      v2f a3 = {h3p[0], h3p[1]};
      const float* wc = w2blk + (size_t)(kk + ko) * D_ + n0 + lm;
      v2f b0  = {wc[0], wc[D_]};
      v2f b1v = {wc[16 + 0], wc[16 + D_]};
      v2f b2v = {wc[32 + 0], wc[32 + D_]};
      v2f b3v = {wc[48 + 0], wc[48 + D_]};
      v2f av[4] = {a0, a1, a2, a3};
      v2f bv[4] = {b0, b1v, b2v, b3v};
#pragma unroll
      for (int mt = 0; mt < 4; ++mt)
#pragma unroll
        for (int nt = 0; nt < 4; ++nt)
          acc2[mt][nt] = __builtin_amdgcn_wmma_f32_16x16x4_f32(
              false, av[mt], false, bv[nt], (short)0, acc2[mt][nt], false,
              false);
    }
  }

  // ---- Epilogue: + b2, * gate, scatter-accumulate into out ----------------
  const float* b2e = b2 + (size_t)e * D_;
#pragma unroll
  for (int mt = 0; mt < 4; ++mt) {
#pragma unroll
    for (int r = 0; r < 8; ++r) {
      int row = mt * 16 + moff + r;
      if (row < rows) {
        float g = sGate[row];
        size_t obase = (size_t)sTok[row] * D_;
#pragma unroll
        for (int nt = 0; nt < 4; ++nt) {
          int col = n0 + nt * 16 + lm;
          float v = (acc2[mt][nt][r] + b2e[col]) * g;
          unsafeAtomicAdd(out + obase + col, v);
        }
      }
    }
  }
}

// ---------------------------------------------------------------------------
extern "C" void kernel_launch(void* const* d_in, const int* in_sizes, int n_in,
                              void* d_out, int out_size, void* d_ws,
                              size_t ws_size, hipStream_t stream) {
  const float* x  = (const float*)d_in[0];
  const float* rw = (const float*)d_in[1];
  const float* rb = (const float*)d_in[2];
  const float* w1 = (const float*)d_in[3];
  const float* b1 = (const float*)d_in[4];
  const float* w2 = (const float*)d_in[5];
  const float* b2 = (const float*)d_in[6];
  float* out = (float*)d_out;

  const int T = in_sizes[0] / D_;  // B*S tokens

  // Workspace layout: [counts: E ints][lists: E*T ints][gates: E*T floats]
  int* counts = (int*)d_ws;
  int* lists  = (int*)((char*)d_ws + 256);
  float* gates = (float*)((char*)d_ws + 256 + (size_t)E_ * T * sizeof(int));

  hipMemsetAsync(counts, 0, E_ * sizeof(int), stream);
  hipMemsetAsync(out, 0, (size_t)out_size * sizeof(float), stream);

  // Router: 8 waves (tokens) per 256-thread block.
  moe_router_kernel<<<(T + 7) / 8, 256, 0, stream>>>(x, rw, rb, T, counts,
                                                     lists, gates);

  // Expert MLP: worst-case tiles per expert = T/MT; inactive tiles exit early.
  dim3 grid((T + MT - 1) / MT, E_);
  size_t smem = (size_t)(MT * XS + MT * HS) * sizeof(float);
  moe_expert_kernel<<<grid, NTHREADS, smem, stream>>>(
      x, w1, b1, w2, b2, counts, lists, gates, out, T);
}